// DecodeModel_25735444038103
// MI455X (gfx1250) — compile-verified
//
#include <hip/hip_runtime.h>
#include <hip/hip_bf16.h>
#include <stddef.h>

// ---------------- GPT-2 decode-step config (matches reference) ----------------
#define Bc  16
#define Sc  1023
#define S1c 1024
#define Lc  12
#define Dc  768
#define Hc  12
#define HDc 64          // D/H
#define Fc  3072        // 4*D
#define Vc  50257
#define Pc  1024
#define EPSc 1e-5f

typedef __attribute__((ext_vector_type(2))) float v2f;
typedef __attribute__((ext_vector_type(8))) float v8f;

__device__ __forceinline__ float gelu_new(float x) {
    // jax.nn.gelu(approximate=True): 0.5*x*(1+tanh(sqrt(2/pi)*(x+0.044715 x^3)))
    float x3 = x * x * x;
    return 0.5f * x * (1.0f + tanhf(0.7978845608028654f * (x + 0.044715f * x3)));
}

// ---------------------------------------------------------------------------
// last_pos[b] = sum(attention_mask[b,:]) - 1
// ---------------------------------------------------------------------------
__global__ void lastpos_kernel(const int* __restrict__ mask, int* __restrict__ last_pos) {
    __shared__ int red[256];
    int b = blockIdx.x, t = threadIdx.x;
    int s = 0;
    for (int i = t; i < S1c; i += 256) s += mask[b * S1c + i];
    red[t] = s; __syncthreads();
    for (int off = 128; off > 0; off >>= 1) {
        if (t < off) red[t] += red[t + off];
        __syncthreads();
    }
    if (t == 0) last_pos[b] = red[0] - 1;
}

// ---------------------------------------------------------------------------
// hs[b,:] = wte[input_ids[b],:] + wpe[0,:]
// ---------------------------------------------------------------------------
__global__ void embed_kernel(const int* __restrict__ ids, const float* __restrict__ wte,
                             const float* __restrict__ wpe, float* __restrict__ hs) {
    int b = blockIdx.x;
    int id = ids[b];
    for (int d = threadIdx.x; d < Dc; d += blockDim.x)
        hs[b * Dc + d] = wte[(size_t)id * Dc + d] + wpe[d];
}

// ---------------------------------------------------------------------------
// LayerNorm over last dim (D), one block per batch row
// ---------------------------------------------------------------------------
__global__ void ln_kernel(const float* __restrict__ in, const float* __restrict__ w,
                          const float* __restrict__ bb, float* __restrict__ out) {
    __shared__ float red[256];
    int b = blockIdx.x, t = threadIdx.x;
    const float* x = in + (size_t)b * Dc;
    float s = 0.0f;
    for (int d = t; d < Dc; d += 256) s += x[d];
    red[t] = s; __syncthreads();
    for (int off = 128; off > 0; off >>= 1) { if (t < off) red[t] += red[t + off]; __syncthreads(); }
    float mu = red[0] * (1.0f / Dc);
    __syncthreads();
    float v = 0.0f;
    for (int d = t; d < Dc; d += 256) { float u = x[d] - mu; v += u * u; }
    red[t] = v; __syncthreads();
    for (int off = 128; off > 0; off >>= 1) { if (t < off) red[t] += red[t + off]; __syncthreads(); }
    float rstd = rsqrtf(red[0] * (1.0f / Dc) + EPSc);
    for (int d = t; d < Dc; d += 256)
        out[(size_t)b * Dc + d] = (x[d] - mu) * rstd * w[d] + bb[d];
}

// ---------------------------------------------------------------------------
// f32 WMMA GEMM: out[16,N] = act(X[16,K] @ W + bias) (+ resid)
//   TRANSB=false : W is [K,N] row-major  (B[k][n] = W[k*N+n])
//   TRANSB=true  : W is [N,K] row-major  (B[k][n] = W[n*K+k])  -> logits @ wte^T
// One wave32 owns one 16x16 output tile; K-loop steps of 4 using
// V_WMMA_F32_16X16X4_F32. Fragment layouts per ISA 7.12.2:
//   A 16x4 : lane half picks K base {0,2}; vgpr picks K parity. a.x=K(2h), a.y=K(2h+1)
//   B 4x16 : N = lane%16, same K mapping as A
//   C/D    : acc[r] = D[M = r + 8*half][N = lane%16]
// EXEC stays all-1s through every WMMA (loads use clamped indices; divergent
// exit happens only after the K-loop).
// ---------------------------------------------------------------------------
template <int ACT, bool TRANSB>
__global__ void gemm16_kernel(const float* __restrict__ X, const float* __restrict__ W,
                              const float* __restrict__ bias, const float* __restrict__ resid,
                              float* __restrict__ out, int K, int N) {
    int lane = threadIdx.x & 31;
    int wid  = threadIdx.x >> 5;
    int tile = blockIdx.x * (blockDim.x >> 5) + wid;
    int half = lane >> 4;
    int lm   = lane & 15;
    int n0   = tile * 16;
    int n    = n0 + lm;
    int nl   = (n < N) ? n : (N - 1);   // clamp so loads never diverge

    const float* xrow = X + (size_t)lm * K + 2 * half;
    v8f acc = {};

    if (!TRANSB) {
        const float* wcol = W + nl;
        for (int k = 0; k < K; k += 4) {
            v2f a, b;
            a.x = xrow[k];
            a.y = xrow[k + 1];
            size_t kk = (size_t)(k + 2 * half);
            b.x = wcol[kk * N];
            b.y = wcol[(kk + 1) * N];
            acc = __builtin_amdgcn_wmma_f32_16x16x4_f32(
                false, a, false, b, (short)0, acc, false, false);
        }
    } else {
        const float* wrow = W + (size_t)nl * K + 2 * half;
        for (int k = 0; k < K; k += 4) {
            v2f a, b;
            a.x = xrow[k];
            a.y = xrow[k + 1];
            b.x = wrow[k];
            b.y = wrow[k + 1];
            acc = __builtin_amdgcn_wmma_f32_16x16x4_f32(
                false, a, false, b, (short)0, acc, false, false);
        }
    }

    if (n >= N) return;  // tail tiles of the vocab GEMM
    float bv = bias ? bias[n] : 0.0f;
#pragma unroll
    for (int r = 0; r < 8; ++r) {
        int m = r + 8 * half;
        float v = acc[r] + bv;
        if (ACT == 1) v = gelu_new(v);
        if (resid) v += resid[(size_t)m * N + n];
        out[(size_t)m * N + n] = v;
    }
}

// ---------------------------------------------------------------------------
// KV append: out[b,s,:] = (s==last_pos[b]) ? new_kv[b,:] : (s<S ? cache[b,s,:] : 0)
// Pure float4 streaming straight into d_out (memory-bound path; ~100MB/layer).
// ---------------------------------------------------------------------------
__global__ void kv_append_kernel(const float* __restrict__ cache,  // [B,S,D] (layer slice)
                                 const float* __restrict__ qkv,    // [B,3D]
                                 int qkv_off,                      // D for k, 2D for v
                                 const int* __restrict__ last_pos,
                                 float* __restrict__ outKV) {      // [B,S1,D]
    const int D4 = Dc / 4;
    size_t idx = (size_t)blockIdx.x * blockDim.x + threadIdx.x;  // over B*S1*D4
    if (idx >= (size_t)Bc * S1c * D4) return;
    int d4 = (int)(idx % D4);
    int s  = (int)((idx / D4) % S1c);
    int b  = (int)(idx / ((size_t)D4 * S1c));
    float4 val;
    if (s == last_pos[b]) {
        const float* src = qkv + (size_t)b * 3 * Dc + qkv_off + d4 * 4;
        val = make_float4(src[0], src[1], src[2], src[3]);
    } else if (s < Sc) {
        val = *(const float4*)(cache + ((size_t)b * Sc + s) * Dc + d4 * 4);
    } else {
        val = make_float4(0.f, 0.f, 0.f, 0.f);
    }
    *(float4*)(outKV + ((size_t)b * S1c + s) * Dc + d4 * 4) = val;
}

// ---------------------------------------------------------------------------
// Attention for one (b,h): scores over S1=1024, softmax, weighted V sum.
// 256 threads/block, LDS for q / scores / reductions. K rows streamed with
// float4 loads (coalesced 256B per row), V rows with scalar loads, + explicit prefetch of upcoming rows.
// ---------------------------------------------------------------------------
__global__ void attn_kernel(const float* __restrict__ qkv,   // [B,3D]
                            const float* __restrict__ kbuf,  // [B,S1,D] (layer slice of new_k)
                            const float* __restrict__ vbuf,  // [B,S1,D] (layer slice of new_v)
                            const int* __restrict__ mask,    // [B,S1]
                            const float* __restrict__ brow,  // [P] (layer slice)
                            float* __restrict__ ctx) {       // [B,D]
    __shared__ float qs[HDc];
    __shared__ float sc[S1c];
    __shared__ float red[256];
    __shared__ float part[4][HDc];

    int bh = blockIdx.x;
    int b = bh / Hc, h = bh % Hc;
    int t = threadIdx.x;

    if (t < HDc) qs[t] = qkv[(size_t)b * 3 * Dc + h * HDc + t] * 0.125f;  // q / sqrt(64)
    __syncthreads();

    // scores (4 positions per thread)
#pragma unroll
    for (int j = 0; j < S1c / 256; ++j) {
        int s = t + j * 256;
        float val;
        if (mask[(size_t)b * S1c + s] == 0) {
            val = -1e9f;
        } else {
            const float4* krow = (const float4*)(kbuf + ((size_t)b * S1c + s) * Dc + h * HDc);
            __builtin_prefetch(kbuf + ((size_t)b * S1c + s + 256) * Dc + h * HDc, 0, 1);
            float acc = 0.0f;
#pragma unroll
            for (int i = 0; i < HDc / 4; ++i) {
                float4 k4 = krow[i];
                acc += qs[4*i]   * k4.x + qs[4*i+1] * k4.y
                     + qs[4*i+2] * k4.z + qs[4*i+3] * k4.w;
            }
            val = acc + brow[s];
        }
        sc[s] = val;
    }
    __syncthreads();

    // softmax: max-reduce
    float m = -1e30f;
    for (int s = t; s < S1c; s += 256) m = fmaxf(m, sc[s]);
    red[t] = m; __syncthreads();
    for (int off = 128; off > 0; off >>= 1) { if (t < off) red[t] = fmaxf(red[t], red[t + off]); __syncthreads(); }
    float mx = red[0];
    __syncthreads();

    // exp + sum-reduce
    float sum = 0.0f;
    for (int s = t; s < S1c; s += 256) { float e = __expf(sc[s] - mx); sc[s] = e; sum += e; }
    red[t] = sum; __syncthreads();
    for (int off = 128; off > 0; off >>= 1) { if (t < off) red[t] += red[t + off]; __syncthreads(); }
    float inv = 1.0f / red[0];
    __syncthreads();

    // ctx[d] = sum_s w[s] * V[b,s,h,d] ; 4 groups of 64 threads split s-space
    int d = t & (HDc - 1);
    int g = t >> 6;
    float acc = 0.0f;
    for (int s = g; s < S1c; s += 4) {
        __builtin_prefetch(vbuf + ((size_t)b * S1c + s + 32) * Dc + h * HDc + d, 0, 1);
        acc += sc[s] * vbuf[((size_t)b * S1c + s) * Dc + h * HDc + d];
    }
    part[g][d] = acc; __syncthreads();
    if (t < HDc)
        ctx[(size_t)b * Dc + t + h * HDc - (t) + (h * HDc + t) - (h * HDc)] =  // = b*D + h*HD + t
            (part[0][t] + part[1][t] + part[2][t] + part[3][t]) * inv;
}

// ---------------------------------------------------------------------------
extern "C" void kernel_launch(void* const* d_in, const int* in_sizes, int n_in,
                              void* d_out, int out_size, void* d_ws, size_t ws_size,
                              hipStream_t stream) {
    (void)in_sizes; (void)n_in; (void)out_size; (void)ws_size;

    const int*   ids     = (const int*)d_in[0];
    const int*   amask   = (const int*)d_in[1];
    const float* cache_k = (const float*)d_in[2];
    const float* cache_v = (const float*)d_in[3];
    const float* wte     = (const float*)d_in[4];
    const float* wpe     = (const float*)d_in[5];
    const float* ln1_w   = (const float*)d_in[6];
    const float* ln1_b   = (const float*)d_in[7];
    const float* caw     = (const float*)d_in[8];
    const float* cab     = (const float*)d_in[9];
    const float* apw     = (const float*)d_in[10];
    const float* apb     = (const float*)d_in[11];
    const float* ln2_w   = (const float*)d_in[12];
    const float* ln2_b   = (const float*)d_in[13];
    const float* fcw     = (const float*)d_in[14];
    const float* fcb     = (const float*)d_in[15];
    const float* mpw     = (const float*)d_in[16];
    const float* mpb     = (const float*)d_in[17];
    const float* lnf_w   = (const float*)d_in[18];
    const float* lnf_b   = (const float*)d_in[19];
    const float* abrow   = (const float*)d_in[20];

    // output: logits [B,V] , new_k [L,B,S1,D] , new_v [L,B,S1,D]
    float* out_logits = (float*)d_out;
    float* out_k = out_logits + (size_t)Bc * Vc;
    float* out_v = out_k + (size_t)Lc * Bc * S1c * Dc;

    // workspace carve-up (all f32)
    float* hs  = (float*)d_ws;             // [B,D]
    float* xb  = hs  + Bc * Dc;            // [B,D]
    float* qkv = xb  + Bc * Dc;            // [B,3D]
    float* ctx = qkv + Bc * 3 * Dc;        // [B,D]
    float* ffh = ctx + Bc * Dc;            // [B,F]
    int*   lp  = (int*)(ffh + Bc * Fc);    // [B]

    const int kvGrid = (Bc * S1c * (Dc / 4) + 255) / 256;

    lastpos_kernel<<<Bc, 256, 0, stream>>>(amask, lp);
    embed_kernel<<<Bc, 256, 0, stream>>>(ids, wte, wpe, hs);

    for (int l = 0; l < Lc; ++l) {
        const size_t lBSD = (size_t)l * Bc * S1c * Dc;
        ln_kernel<<<Bc, 256, 0, stream>>>(hs, ln1_w + (size_t)l * Dc, ln1_b + (size_t)l * Dc, xb);
        gemm16_kernel<0, false><<<(3 * Dc) / 64, 128, 0, stream>>>(
            xb, caw + (size_t)l * Dc * 3 * Dc, cab + (size_t)l * 3 * Dc, nullptr, qkv, Dc, 3 * Dc);
        kv_append_kernel<<<kvGrid, 256, 0, stream>>>(
            cache_k + (size_t)l * Bc * Sc * Dc, qkv, Dc, lp, out_k + lBSD);
        kv_append_kernel<<<kvGrid, 256, 0, stream>>>(
            cache_v + (size_t)l * Bc * Sc * Dc, qkv, 2 * Dc, lp, out_v + lBSD);
        attn_kernel<<<Bc * Hc, 256, 0, stream>>>(
            qkv, out_k + lBSD, out_v + lBSD, amask, abrow + (size_t)l * Pc, ctx);
        gemm16_kernel<0, false><<<Dc / 64, 128, 0, stream>>>(
            ctx, apw + (size_t)l * Dc * Dc, apb + (size_t)l * Dc, hs, hs, Dc, Dc);
        ln_kernel<<<Bc, 256, 0, stream>>>(hs, ln2_w + (size_t)l * Dc, ln2_b + (size_t)l * Dc, xb);
        gemm16_kernel<1, false><<<Fc / 64, 128, 0, stream>>>(
            xb, fcw + (size_t)l * Dc * Fc, fcb + (size_t)l * Fc, nullptr, ffh, Dc, Fc);
        gemm16_kernel<0, false><<<Dc / 64, 128, 0, stream>>>(
            ffh, mpw + (size_t)l * Fc * Dc, mpb + (size_t)l * Dc, hs, hs, Fc, Dc);
    }

    // final LN + tied LM head (x @ wte^T), tail-predicated over V
    ln_kernel<<<Bc, 256, 0, stream>>>(hs, lnf_w, lnf_b, xb);
    int vtiles = (Vc + 15) / 16;
    gemm16_kernel<0, true><<<(vtiles + 3) / 4, 128, 0, stream>>>(
        xb, wte, nullptr, nullptr, out_logits, Dc, Vc);
}